// MSAGATNet_30081950941211
// MI455X (gfx1250) — compile-verified
//
#include <hip/hip_runtime.h>
#include <math.h>

// ---- model dims ----
#define BATCH 4
#define WLEN  64
#define MN    2048
#define HIDD  32
#define NHEAD 4
#define LRK   8
#define HORZ  24
#define TCH   16
#define NSC   4
#define KSZ   3
#define EPSV  1e-5f
#define NROWS (BATCH*MN)   // 8192

typedef __attribute__((ext_vector_type(2))) float v2f;
typedef __attribute__((ext_vector_type(8))) float v8f;

// D = A(16x4 f32) * B(4x16 f32) + C(16x16 f32), full-precision WMMA
static __device__ __forceinline__ v8f wmma4(v2f a, v2f b, v8f c) {
  return __builtin_amdgcn_wmma_f32_16x16x4_f32(false, a, false, b, (short)0, c,
                                               false, false);
}

// ============================================================================
// P: collapse low-rank weight pairs, fold BN, softmax(alpha), and pre-pack all
//    WMMA B-operands into per-lane fragment order (zero-padding baked in).
// ============================================================================
__global__ __launch_bounds__(256) void k_prep(
    const float* __restrict__ fp_low_w,
    const float* __restrict__ qkv_low_w, const float* __restrict__ qkv_low_b,
    const float* __restrict__ qkv_high_w, const float* __restrict__ qkv_high_b,
    const float* __restrict__ out_low_w, const float* __restrict__ out_low_b,
    const float* __restrict__ out_high_w, const float* __restrict__ out_high_b,
    const float* __restrict__ fus_low_w, const float* __restrict__ fus_low_b,
    const float* __restrict__ fus_high_w, const float* __restrict__ fus_high_b,
    const float* __restrict__ tconv_w, const float* __restrict__ tbn_g,
    const float* __restrict__ fusion_w,
    float* __restrict__ Wqkv, float* __restrict__ bqkv,
    float* __restrict__ Wout, float* __restrict__ bout,
    float* __restrict__ Wfus, float* __restrict__ bfus,
    float* __restrict__ tgp, float* __restrict__ alphaW,
    float* __restrict__ Wlowf, float* __restrict__ Wqkvf, float* __restrict__ Bmf)
{
  const int tid = threadIdx.x;
  // ---- phase 1: collapsed weights ----
  // Wqkv[32][96] = qkv_low[32,24] @ qkv_high[24,96]
  for (int idx = tid; idx < HIDD * 96; idx += 256) {
    int i = idx / 96, j = idx % 96;
    float s = 0.f;
    for (int r = 0; r < 24; ++r) s = fmaf(qkv_low_w[i*24+r], qkv_high_w[r*96+j], s);
    Wqkv[idx] = s;
  }
  for (int j = tid; j < 96; j += 256) {
    float s = qkv_high_b[j];
    for (int r = 0; r < 24; ++r) s = fmaf(qkv_low_b[r], qkv_high_w[r*96+j], s);
    bqkv[j] = s;
  }
  // Wout[32][32] = out_low[32,8] @ out_high[8,32]
  for (int idx = tid; idx < HIDD * HIDD; idx += 256) {
    int i = idx / 32, j = idx % 32;
    float s = 0.f;
    for (int r = 0; r < 8; ++r) s = fmaf(out_low_w[i*8+r], out_high_w[r*32+j], s);
    Wout[idx] = s;
  }
  for (int j = tid; j < 32; j += 256) {
    float s = out_high_b[j];
    for (int r = 0; r < 8; ++r) s = fmaf(out_low_b[r], out_high_w[r*32+j], s);
    bout[j] = s;
  }
  // Wfus[32][32] = fus_low[32,8] @ fus_high[8,32]
  for (int idx = tid; idx < HIDD * HIDD; idx += 256) {
    int i = idx / 32, j = idx % 32;
    float s = 0.f;
    for (int r = 0; r < 8; ++r) s = fmaf(fus_low_w[i*8+r], fus_high_w[r*32+j], s);
    Wfus[idx] = s;
  }
  for (int j = tid; j < 32; j += 256) {
    float s = fus_high_b[j];
    for (int r = 0; r < 8; ++r) s = fmaf(fus_low_b[r], fus_high_w[r*32+j], s);
    bfus[j] = s;
  }
  for (int idx = tid; idx < NSC * HIDD; idx += 256)
    tgp[idx] = tbn_g[idx] * rsqrtf(1.f + EPSV);
  if (tid == 0) {
    float mx = fusion_w[0];
    for (int i = 1; i < 4; ++i) mx = fmaxf(mx, fusion_w[i]);
    float e[4], se = 0.f;
    for (int i = 0; i < 4; ++i) { e[i] = __expf(fusion_w[i]-mx); se += e[i]; }
    for (int i = 0; i < 4; ++i) alphaW[i] = e[i] / se;
  }
  __syncthreads();
  // ---- phase 2: fragment-order B operands ----
  // Wlowf: [256 k-steps][32 lanes][2]; lane = kh*16+n; zero-pad n>=8.
  for (int idx = tid; idx < 256 * 32; idx += 256) {
    int step = idx >> 5, lane = idx & 31;
    int n = lane & 15, kh = lane >> 4;
    int k = step * 4;
    Wlowf[idx * 2 + 0] = (n < 8) ? fp_low_w[(k + kh) * 8 + n]     : 0.f;
    Wlowf[idx * 2 + 1] = (n < 8) ? fp_low_w[(k + 2 + kh) * 8 + n] : 0.f;
  }
  // Wqkvf: [6 n-tiles * 8 k-steps][32 lanes][2]
  for (int idx = tid; idx < 48 * 32; idx += 256) {
    int step = idx >> 5, lane = idx & 31;
    int nt = step >> 3, ks = step & 7;
    int n = lane & 15, kh = lane >> 4;
    int k = ks * 4;
    Wqkvf[idx * 2 + 0] = Wqkv[(k + kh) * 96 + nt * 16 + n];
    Wqkvf[idx * 2 + 1] = Wqkv[(k + 2 + kh) * 96 + nt * 16 + n];
  }
  // Bmf: [4 scales][24 k-steps][2 n-tiles][32 lanes][2], straight from tconv_w.
  // column kcol = kk*32 + ih  ->  tconv_w[i][o][ih][kk]
  for (int idx = tid; idx < NSC * 24 * 2 * 32; idx += 256) {
    int lane = idx & 31;
    int tt = (idx >> 5) & 1;
    int sstep = (idx >> 6) % 24;
    int i = idx / (24 * 2 * 32);
    int n = lane & 15, kh = lane >> 4;
    int o = tt * 16 + n;
    int kc = sstep * 4;
    int kcol0 = kc + kh, kcol1 = kc + 2 + kh;
    int kk0 = kcol0 >> 5, ih0 = kcol0 & 31;
    int kk1 = kcol1 >> 5, ih1 = kcol1 & 31;
    Bmf[idx * 2 + 0] = tconv_w[(((i*32+o)*32+ih0)*3) + kk0];
    Bmf[idx * 2 + 1] = tconv_w[(((i*32+o)*32+ih1)*3) + kk1];
  }
}

// ============================================================================
// AB: depthwise+pointwise conv fused as A-fragment of [8192x1024]@[1024x8]
//     WMMA GEMM (N padded to 16), fused 8->32 expansion. One wave / 16 rows.
//     Inner step: 1x ds_load_b64 (A taps) + 1x coalesced global b64 (B frag).
// ============================================================================
__global__ __launch_bounds__(32) void k_conv_feat(
    const float* __restrict__ x,
    const float* __restrict__ dw_w, const float* __restrict__ dw_b,
    const float* __restrict__ bn1_g, const float* __restrict__ bn1_b,
    const float* __restrict__ pw_w, const float* __restrict__ pw_b,
    const float* __restrict__ bn2_g, const float* __restrict__ bn2_b,
    const float* __restrict__ Wlowf, const float* __restrict__ fp_low_b,
    const float* __restrict__ fp_high_w, const float* __restrict__ fp_high_b,
    float* __restrict__ feat)
{
  __shared__ float xt[16][64];
  __shared__ float s[16][64];
  __shared__ float s2g[16], s2b[16];
  __shared__ float f8[16][8];
  const int l = threadIdx.x;
  const int r0 = blockIdx.x * 16;
  const int bb = r0 >> 11;
  const int mb = r0 & (MN - 1);
  if (l < 16) {   // fold pointwise conv + bn2 into per-channel scale/bias
    float gg = bn2_g[l] * rsqrtf(1.f + EPSV);
    s2g[l] = gg * pw_w[l];
    s2b[l] = gg * pw_b[l] + bn2_b[l];
  }
  // coalesced x tile load: lane -> node index (consecutive m), t slow
  const float* xb = x + (size_t)bb * WLEN * MN + mb;
  for (int p = l; p < 16 * 64; p += 32) {
    int rr = p & 15, t = p >> 4;
    xt[rr][t] = xb[(size_t)t * MN + rr];
  }
  __syncthreads();
  const float g1 = bn1_g[0] * rsqrtf(1.f + EPSV);
  const float w0 = dw_w[0], w1 = dw_w[1], w2 = dw_w[2];
  const float db = dw_b[0], bb1 = bn1_b[0];
  for (int p = l; p < 16 * 64; p += 32) {
    int rr = p & 15, t = p >> 4;
    float xm1 = (t > 0)  ? xt[rr][t-1] : 0.f;
    float xc  = xt[rr][t];
    float xp1 = (t < 63) ? xt[rr][t+1] : 0.f;
    float h = fmaf(w0, xm1, fmaf(w1, xc, fmaf(w2, xp1, db)));
    s[rr][t] = fmaxf(fmaf(h, g1, bb1), 0.f);
  }
  __syncthreads();
  const int row = l & 15, kh = l >> 4, n = l & 15;
  v8f acc = {};
  for (int c = 0; c < TCH; ++c) {          // channel-outer: scale/bias hoisted
    const float sg = s2g[c], sb = s2b[c];
    const float* wf = Wlowf + (size_t)(c * 16) * 64;
#pragma unroll
    for (int kt = 0; kt < 16; ++kt) {
      int t0 = kt * 4 + 2 * kh;            // even -> aligned pair, same channel
      v2f sv = *(const v2f*)&s[row][t0];
      v2f a;
      a.x = fmaxf(fmaf(sg, sv.x, sb), 0.f);
      a.y = fmaxf(fmaf(sg, sv.y, sb), 0.f);
      v2f bv = *(const v2f*)(wf + (kt * 32 + l) * 2);
      acc = wmma4(a, bv, acc);
    }
  }
  if (n < 8) {
    float bl = fp_low_b[n];
#pragma unroll
    for (int j = 0; j < 8; ++j) f8[j + 8 * kh][n] = acc[j] + bl;
  }
  __syncthreads();
  // fused 8->32 expansion
#pragma unroll
  for (int o16 = 0; o16 < 16; ++o16) {
    int o = kh * 16 + o16;
    float v = fp_high_b[o];
#pragma unroll
    for (int r2 = 0; r2 < 8; ++r2) v = fmaf(f8[row][r2], fp_high_w[r2 * 32 + o], v);
    feat[(size_t)(r0 + row) * HIDD + o] = v;
  }
}

// ============================================================================
// D1: qkv = feat[8192x32] @ Wqkv[32x96] + bqkv   (WMMA, 6 N-tiles, K=32)
// ============================================================================
__global__ __launch_bounds__(32) void k_qkv(
    const float* __restrict__ feat, const float* __restrict__ Wqkvf,
    const float* __restrict__ bqkv, float* __restrict__ qkv)
{
  __shared__ float As[16][32];
  const int l = threadIdx.x;
  const int r0 = blockIdx.x * 16;
  for (int p = l; p < 512; p += 32)
    As[p >> 5][p & 31] = feat[(size_t)r0 * HIDD + p];   // contiguous, coalesced
  __syncthreads();
  const int row = l & 15, kh = l >> 4, n = l & 15;
#pragma unroll
  for (int nt = 0; nt < 6; ++nt) {
    v8f acc = {};
#pragma unroll
    for (int ks = 0; ks < 8; ++ks) {
      v2f a = *(const v2f*)&As[row][ks * 4 + 2 * kh];
      v2f bv = *(const v2f*)(Wqkvf + (size_t)((nt * 8 + ks) * 32 + l) * 2);
      acc = wmma4(a, bv, acc);
    }
    float bl = bqkv[nt * 16 + n];
#pragma unroll
    for (int j = 0; j < 8; ++j)
      qkv[(size_t)(r0 + j + 8 * kh) * 96 + nt * 16 + n] = acc[j] + bl;
  }
}

// ============================================================================
// D2: per (b,h) linear-attention kv[8][8] = (elu(k)+1)^T @ v over N=2048.
//     Lane owns (d,e) pairs; 8 waves strided over N; one LDS pass to reduce.
// ============================================================================
__global__ __launch_bounds__(256) void k_kv(
    const float* __restrict__ qkv, float* __restrict__ kv)
{
  const int tid = threadIdx.x;
  const int lane = tid & 31, w = tid >> 5;
  const int bh = blockIdx.x;          // b*4 + h
  const int b = bh >> 2, h = bh & 3;
  const int d0 = lane >> 3, e0 = lane & 7;
  float acc0 = 0.f, acc1 = 0.f;
  for (int nn = w; nn < MN; nn += 8) {
    const float* base = qkv + (size_t)(b * MN + nn) * 96;
    float k0 = base[32 + h * 8 + d0];
    float k1 = base[32 + h * 8 + 4 + d0];
    float ve = base[64 + h * 8 + e0];
    float kf0 = (k0 > 0.f) ? (k0 + 1.f) : __expf(k0);
    float kf1 = (k1 > 0.f) ? (k1 + 1.f) : __expf(k1);
    acc0 = fmaf(kf0, ve, acc0);
    acc1 = fmaf(kf1, ve, acc1);
  }
  __shared__ float red[8][64];
  red[w][lane] = acc0;        // entry d0*8+e0  == lane
  red[w][32 + lane] = acc1;   // entry (4+d0)*8+e0 == 32+lane
  __syncthreads();
  if (tid < 64) {
    float s = 0.f;
#pragma unroll
    for (int w2 = 0; w2 < 8; ++w2) s += red[w2][tid];
    kv[bh * 64 + tid] = s;
  }
}

// ============================================================================
// D3: linear-attn output + collapsed out-projection + residual
// ============================================================================
__global__ __launch_bounds__(256) void k_attn_out(
    const float* __restrict__ qkv, const float* __restrict__ kv,
    const float* __restrict__ Wout, const float* __restrict__ bout,
    const float* __restrict__ feat, float* __restrict__ feat2)
{
  const int r = blockIdx.x * 256 + threadIdx.x;
  const int b = r >> 11;
  const float* qk = qkv + (size_t)r * 96;
  float attn[32];
#pragma unroll
  for (int h = 0; h < 4; ++h) {
    const float* kvh = kv + (size_t)(b * 4 + h) * 64;
    float qf[8]; float ks = 0.f;
#pragma unroll
    for (int d = 0; d < 8; ++d) {
      float qr = qk[h * 8 + d];
      qf[d] = (qr > 0.f) ? (qr + 1.f) : __expf(qr);
      float kr = qk[32 + h * 8 + d];
      ks += (kr > 0.f) ? (kr + 1.f) : __expf(kr);
    }
    float z = 1.f / (ks + 1e-8f);
#pragma unroll
    for (int e = 0; e < 8; ++e) {
      float sm = 0.f;
#pragma unroll
      for (int d = 0; d < 8; ++d) sm = fmaf(qf[d], kvh[d * 8 + e], sm);
      attn[h * 8 + e] = sm * z;
    }
  }
  for (int o = 0; o < 32; ++o) {
    float v = bout[o];
#pragma unroll
    for (int i = 0; i < 32; ++i) v = fmaf(attn[i], Wout[i * 32 + o], v);
    feat2[(size_t)r * 32 + o] = feat[(size_t)r * 32 + o] + v;
  }
}

// ============================================================================
// F: dilated multi-scale node-axis convs as implicit WMMA GEMM with fused
//    BN/ReLU/alpha, + fusion projection + residual + LayerNorm
// ============================================================================
__global__ __launch_bounds__(32) void k_temporal(
    const float* __restrict__ feat2, const float* __restrict__ Bmf,
    const float* __restrict__ tconv_b, const float* __restrict__ tgp,
    const float* __restrict__ tbn_b, const float* __restrict__ alphaW,
    const float* __restrict__ Wfus, const float* __restrict__ bfus,
    const float* __restrict__ ln_g, const float* __restrict__ ln_b,
    float* __restrict__ feat3)
{
  __shared__ float As[32][32];    // halo window: nodes m0-8 .. m0+23, 32 ch
  __shared__ float tmp[16][32];
  __shared__ float tmp2[16][32];
  const int l = threadIdx.x;
  const int blk = blockIdx.x;
  const int b = blk >> 7;               // 128 tiles per batch
  const int m0 = (blk & 127) * 16;
  for (int p = l; p < 32 * 32; p += 32) {
    int rr = p >> 5, ch = p & 31;
    int mm = m0 - 8 + rr;
    As[rr][ch] = (mm >= 0 && mm < MN) ? feat2[(size_t)(b * MN + mm) * 32 + ch] : 0.f;
  }
  __syncthreads();
  const int row = l & 15, kh = l >> 4, n = l & 15;
  v8f fus0 = {}, fus1 = {};
#pragma unroll
  for (int i = 0; i < NSC; ++i) {
    const int d = 1 << i;
    v8f a0 = {}, a1 = {};
    for (int sstep = 0; sstep < 24; ++sstep) {
      int ka = sstep * 4 + 2 * kh;
      int kk0 = ka >> 5, ih0 = ka & 31;           // ih0 even, pair in same tap
      v2f a = *(const v2f*)&As[row + (kk0 - 1) * d + 8][ih0];
      const float* bp = Bmf + (size_t)((i * 24 + sstep) * 2) * 64 + l * 2;
      v2f b0 = *(const v2f*)bp;
      v2f b1 = *(const v2f*)(bp + 64);
      a0 = wmma4(a, b0, a0);
      a1 = wmma4(a, b1, a1);
    }
    float al = alphaW[i];
    int o0 = n, o1 = 16 + n;
    float cb0 = tconv_b[i*32+o0], g0  = tgp[i*32+o0], tb0 = tbn_b[i*32+o0];
    float cb1 = tconv_b[i*32+o1], g1v = tgp[i*32+o1], tb1 = tbn_b[i*32+o1];
#pragma unroll
    for (int j = 0; j < 8; ++j) {
      float f0 = fmaxf(fmaf(a0[j] + cb0, g0,  tb0), 0.f);
      float f1 = fmaxf(fmaf(a1[j] + cb1, g1v, tb1), 0.f);
      fus0[j] = fmaf(al, f0, fus0[j]);
      fus1[j] = fmaf(al, f1, fus1[j]);
    }
  }
#pragma unroll
  for (int j = 0; j < 8; ++j) {
    tmp[j + 8 * kh][n]      = fus0[j];
    tmp[j + 8 * kh][16 + n] = fus1[j];
  }
  __syncthreads();
  // fusion projection + residual
  float tv[16];
#pragma unroll
  for (int o16 = 0; o16 < 16; ++o16) {
    int o = kh * 16 + o16;
    float v = bfus[o];
#pragma unroll
    for (int ih = 0; ih < 32; ++ih) v = fmaf(tmp[row][ih], Wfus[ih * 32 + o], v);
    tv[o16] = v + tmp[row][o];
  }
#pragma unroll
  for (int o16 = 0; o16 < 16; ++o16) tmp2[row][kh * 16 + o16] = tv[o16];
  __syncthreads();
  if (l < 16) {
    float mu = 0.f;
#pragma unroll
    for (int o = 0; o < 32; ++o) mu += tmp2[l][o];
    mu *= (1.f / 32.f);
    float var = 0.f;
#pragma unroll
    for (int o = 0; o < 32; ++o) { float dd = tmp2[l][o] - mu; var = fmaf(dd, dd, var); }
    var *= (1.f / 32.f);
    float inv = rsqrtf(var + EPSV);
#pragma unroll
    for (int o = 0; o < 32; ++o)
      feat3[(size_t)(b * MN + m0 + l) * 32 + o] =
          (tmp2[l][o] - mu) * inv * ln_g[o] + ln_b[o];
  }
}

// ============================================================================
// G: progressive prediction head (+ constant attn_reg: softmax rows sum to 1,
//    so mean|attn| == 1/N exactly -> attn_reg = 1e-5/2048)
// ============================================================================
__global__ __launch_bounds__(256) void k_pred(
    const float* __restrict__ x, const float* __restrict__ feat3,
    const float* __restrict__ pred_low_w, const float* __restrict__ pred_low_b,
    const float* __restrict__ pln_g, const float* __restrict__ pln_b,
    const float* __restrict__ pred_high_w, const float* __restrict__ pred_high_b,
    const float* __restrict__ g1_w, const float* __restrict__ g1_b,
    const float* __restrict__ g2_w, const float* __restrict__ g2_b,
    float* __restrict__ out)
{
  const int r = blockIdx.x * 256 + threadIdx.x;
  const int b = r >> 11, m = r & (MN - 1);
  float f[32];
#pragma unroll
  for (int o = 0; o < 32; ++o) f[o] = feat3[(size_t)r * 32 + o];
  float xl[8], gg[8];
#pragma unroll
  for (int j = 0; j < 8; ++j) {
    float s = pred_low_b[j], s2 = g1_b[j];
#pragma unroll
    for (int i = 0; i < 32; ++i) {
      s  = fmaf(f[i], pred_low_w[i * 8 + j], s);
      s2 = fmaf(f[i], g1_w[i * 8 + j], s2);
    }
    xl[j] = s;
    gg[j] = fmaxf(s2, 0.f);
  }
  float mu = 0.f;
#pragma unroll
  for (int j = 0; j < 8; ++j) mu += xl[j];
  mu *= 0.125f;
  float var = 0.f;
#pragma unroll
  for (int j = 0; j < 8; ++j) { float dd = xl[j] - mu; var = fmaf(dd, dd, var); }
  var *= 0.125f;
  float inv = rsqrtf(var + EPSV);
  float xm[8];
#pragma unroll
  for (int j = 0; j < 8; ++j)
    xm[j] = fmaxf((xl[j] - mu) * inv * pln_g[j] + pln_b[j], 0.f);
  float last = x[((size_t)b * WLEN + (WLEN - 1)) * MN + m];
  for (int hh = 0; hh < HORZ; ++hh) {
    float ip = pred_high_b[hh], gt = g2_b[hh];
#pragma unroll
    for (int j = 0; j < 8; ++j) {
      ip = fmaf(xm[j], pred_high_w[j * HORZ + hh], ip);
      gt = fmaf(gg[j], g2_w[j * HORZ + hh], gt);
    }
    float gate = 1.f / (1.f + __expf(-gt));
    float prog = last * __expf(-0.1f * (float)(hh + 1));
    out[(size_t)r * HORZ + hh] = gate * ip + (1.f - gate) * prog;
  }
  if (r == 0) out[(size_t)NROWS * HORZ] = 1e-5f / 2048.f;  // attn_reg (exact)
}

// ============================================================================
extern "C" void kernel_launch(void* const* d_in, const int* in_sizes, int n_in,
                              void* d_out, int out_size, void* d_ws, size_t ws_size,
                              hipStream_t stream) {
  const float* x           = (const float*)d_in[0];
  const float* dw_w        = (const float*)d_in[1];
  const float* dw_b        = (const float*)d_in[2];
  const float* bn1_g       = (const float*)d_in[3];
  const float* bn1_b       = (const float*)d_in[4];
  const float* pw_w        = (const float*)d_in[5];
  const float* pw_b        = (const float*)d_in[6];
  const float* bn2_g       = (const float*)d_in[7];
  const float* bn2_b       = (const float*)d_in[8];
  const float* fp_low_w    = (const float*)d_in[9];
  const float* fp_low_b    = (const float*)d_in[10];
  const float* fp_high_w   = (const float*)d_in[11];
  const float* fp_high_b   = (const float*)d_in[12];
  const float* qkv_low_w   = (const float*)d_in[13];
  const float* qkv_low_b   = (const float*)d_in[14];
  const float* qkv_high_w  = (const float*)d_in[15];
  const float* qkv_high_b  = (const float*)d_in[16];
  const float* out_low_w   = (const float*)d_in[17];
  const float* out_low_b   = (const float*)d_in[18];
  const float* out_high_w  = (const float*)d_in[19];
  const float* out_high_b  = (const float*)d_in[20];
  // d_in[21] adj_u, d_in[22] adj_v: dead (attn_reg is analytically constant)
  const float* tconv_w     = (const float*)d_in[23];
  const float* tconv_b     = (const float*)d_in[24];
  const float* tbn_g       = (const float*)d_in[25];
  const float* tbn_b       = (const float*)d_in[26];
  const float* fusion_w    = (const float*)d_in[27];
  const float* fus_low_w   = (const float*)d_in[28];
  const float* fus_low_b   = (const float*)d_in[29];
  const float* fus_high_w  = (const float*)d_in[30];
  const float* fus_high_b  = (const float*)d_in[31];
  const float* ln_g        = (const float*)d_in[32];
  const float* ln_b        = (const float*)d_in[33];
  const float* pred_low_w  = (const float*)d_in[34];
  const float* pred_low_b  = (const float*)d_in[35];
  const float* pln_g       = (const float*)d_in[36];
  const float* pln_b       = (const float*)d_in[37];
  const float* pred_high_w = (const float*)d_in[38];
  const float* pred_high_b = (const float*)d_in[39];
  const float* g1_w        = (const float*)d_in[40];
  const float* g1_b        = (const float*)d_in[41];
  const float* g2_w        = (const float*)d_in[42];
  const float* g2_b        = (const float*)d_in[43];

  float* ws     = (float*)d_ws;
  float* feat   = ws;                              // 8192*32
  float* qkv    = feat   + (size_t)NROWS * HIDD;   // 8192*96
  float* kvb    = qkv    + (size_t)NROWS * 96;     // 16*64
  float* feat2  = kvb    + (size_t)BATCH * NHEAD * 64;
  float* feat3  = feat2  + (size_t)NROWS * HIDD;
  float* Wqkv   = feat3  + (size_t)NROWS * HIDD;   // 32*96
  float* bqkv   = Wqkv   + 32 * 96;                // 96
  float* Wout   = bqkv   + 96;                     // 32*32
  float* bout   = Wout   + 32 * 32;                // 32
  float* Wfus   = bout   + 32;                     // 32*32
  float* bfus   = Wfus   + 32 * 32;                // 32
  float* tgp    = bfus   + 32;                     // 4*32
  float* alphaW = tgp    + NSC * 32;               // 4
  float* Wlowf  = alphaW + 4;                      // 256*32*2 = 16384
  float* Wqkvf  = Wlowf  + 256 * 32 * 2;           // 48*32*2  = 3072
  float* Bmf    = Wqkvf  + 48 * 32 * 2;            // 4*24*2*32*2 = 12288

  k_prep<<<1, 256, 0, stream>>>(
      fp_low_w,
      qkv_low_w, qkv_low_b, qkv_high_w, qkv_high_b,
      out_low_w, out_low_b, out_high_w, out_high_b,
      fus_low_w, fus_low_b, fus_high_w, fus_high_b,
      tconv_w, tbn_g, fusion_w,
      Wqkv, bqkv, Wout, bout, Wfus, bfus, tgp, alphaW,
      Wlowf, Wqkvf, Bmf);

  k_conv_feat<<<NROWS / 16, 32, 0, stream>>>(
      x, dw_w, dw_b, bn1_g, bn1_b, pw_w, pw_b, bn2_g, bn2_b,
      Wlowf, fp_low_b, fp_high_w, fp_high_b, feat);

  k_qkv<<<NROWS / 16, 32, 0, stream>>>(feat, Wqkvf, bqkv, qkv);

  k_kv<<<BATCH * NHEAD, 256, 0, stream>>>(qkv, kvb);

  k_attn_out<<<NROWS / 256, 256, 0, stream>>>(qkv, kvb, Wout, bout, feat, feat2);

  k_temporal<<<NROWS / 16, 32, 0, stream>>>(
      feat2, Bmf, tconv_b, tgp, tbn_b, alphaW, Wfus, bfus, ln_g, ln_b, feat3);

  k_pred<<<NROWS / 256, 256, 0, stream>>>(
      x, feat3, pred_low_w, pred_low_b, pln_g, pln_b,
      pred_high_w, pred_high_b, g1_w, g1_b, g2_w, g2_b, (float*)d_out);
}